// FaultGAT_83786222010864
// MI455X (gfx1250) — compile-verified
//
#include <hip/hip_runtime.h>
#include <math.h>

typedef __attribute__((ext_vector_type(2))) float v2f;
typedef __attribute__((ext_vector_type(8))) float v8f;

// ---------------------------------------------------------------------------
// fp32 WMMA GEMM: C[nrows, ncols] = A[nrows, K] @ B[K, ncols]
// One wave (32 lanes) per 16x16 output tile, K-loop of V_WMMA_F32_16X16X4_F32.
// nrows, ncols multiples of 16; K multiple of 4. EXEC is all-1s per wave
// (tile-out-of-range exit is wave-uniform).
// A 16x4 frag:  lane m=lane&15 holds A[m][kb], A[m][kb+1], kb=(lane>>4)*2
// B 4x16 frag:  lane n=lane&15 holds B[kb][n], B[kb+1][n]
// C/D 16x16:    VGPR i holds row (i + 8*(lane>>4)), col lane&15
// ---------------------------------------------------------------------------
__global__ void gat_wmma_gemm_f32(const float* __restrict__ A,
                                  const float* __restrict__ B,
                                  float* __restrict__ C,
                                  int nrows, int K, int ncols) {
  int wavesPerBlock = blockDim.x >> 5;
  int wave = blockIdx.x * wavesPerBlock + (threadIdx.x >> 5);
  int lane = threadIdx.x & 31;
  int colTiles = ncols >> 4;
  int rowTiles = nrows >> 4;
  if (wave >= colTiles * rowTiles) return;  // wave-uniform
  int tm = wave / colTiles;
  int tn = wave - tm * colTiles;

  int mrow  = (tm << 4) + (lane & 15);
  int ncol  = (tn << 4) + (lane & 15);
  int khalf = (lane >> 4) << 1;  // 0 or 2

  const float* aRow = A + (size_t)mrow * K;
  v8f acc = {};
  for (int k0 = 0; k0 < K; k0 += 4) {
    int ka = k0 + khalf;
    v2f a, b;
    a.x = aRow[ka];
    a.y = aRow[ka + 1];
    b.x = B[(size_t)ka * ncols + ncol];
    b.y = B[(size_t)(ka + 1) * ncols + ncol];
    acc = __builtin_amdgcn_wmma_f32_16x16x4_f32(
        /*neg_a=*/false, a, /*neg_b=*/false, b,
        /*c_mod=*/(short)0, acc, /*reuse_a=*/false, /*reuse_b=*/false);
  }
  int mbase = (tm << 4) + ((lane >> 4) << 3);
#pragma unroll
  for (int i = 0; i < 8; ++i)
    C[(size_t)(mbase + i) * ncols + ncol] = acc[i];
}

// ---------------------------------------------------------------------------
// Per-(node,head) attention scores: s = <h[n,hd,:], att[hd,:]>
// ---------------------------------------------------------------------------
__global__ void gat_attn_scores(const float* __restrict__ h,
                                const float* __restrict__ a_src,
                                const float* __restrict__ a_dst,
                                float* __restrict__ s_src,
                                float* __restrict__ s_dst,
                                int n, int heads, int ch) {
  int idx = blockIdx.x * blockDim.x + threadIdx.x;
  if (idx >= n * heads) return;
  int node = idx / heads;
  int hd = idx - node * heads;
  const float* hp = h + (size_t)node * heads * ch + (size_t)hd * ch;
  const float* as = a_src + hd * ch;
  const float* ad = a_dst + hd * ch;
  float ss = 0.f, sd = 0.f;
  for (int c = 0; c < ch; ++c) {
    float v = hp[c];
    ss += v * as[c];
    sd += v * ad[c];
  }
  s_src[idx] = ss;
  s_dst[idx] = sd;
}

__global__ void gat_fill_f32(float* __restrict__ p, float v, size_t n) {
  size_t i = (size_t)blockIdx.x * blockDim.x + threadIdx.x;
  size_t stride = (size_t)gridDim.x * blockDim.x;
  for (; i < n; i += stride) p[i] = v;
}

// float atomic-max via int/uint ordering trick (target init = -inf)
__device__ __forceinline__ void atomicMaxF32(float* addr, float val) {
  if (val >= 0.f)
    atomicMax((int*)addr, __float_as_int(val));
  else
    atomicMin((unsigned int*)addr, __float_as_uint(val));
}

// Pass A: e = leaky_relu(s_src[src]+s_dst[dst]); running max per (dst,head)
__global__ void gat_edge_scores(const int* __restrict__ src,
                                const int* __restrict__ dst, int E, int n,
                                const float* __restrict__ s_src,
                                const float* __restrict__ s_dst, int heads,
                                float* __restrict__ escore,
                                float* __restrict__ mmax) {
  int e = blockIdx.x * blockDim.x + threadIdx.x;
  if (e >= E + n) return;
  int s = (e < E) ? src[e] : (e - E);
  int d = (e < E) ? dst[e] : (e - E);
  for (int hd = 0; hd < heads; ++hd) {
    float v = s_src[s * heads + hd] + s_dst[d * heads + hd];
    v = v > 0.f ? v : 0.2f * v;
    escore[(size_t)e * heads + hd] = v;
    atomicMaxF32(&mmax[d * heads + hd], v);
  }
}

// Pass B: ex = exp(e - m[dst]); den[dst] += ex
__global__ void gat_edge_exp(const int* __restrict__ src,
                             const int* __restrict__ dst, int E, int n,
                             int heads, float* __restrict__ escore,
                             const float* __restrict__ mmax,
                             float* __restrict__ den) {
  int e = blockIdx.x * blockDim.x + threadIdx.x;
  if (e >= E + n) return;
  int d = (e < E) ? dst[e] : (e - E);
  for (int hd = 0; hd < heads; ++hd) {
    float ex = __expf(escore[(size_t)e * heads + hd] - mmax[d * heads + hd]);
    escore[(size_t)e * heads + hd] = ex;
    atomicAdd(&den[d * heads + hd], ex);
  }
}

// Pass C (layer 1, heads=4, ch=64): one wave per edge, 2 channels/lane/head
__global__ void gat_aggregate1(const int* __restrict__ src,
                               const int* __restrict__ dst, int E, int n,
                               const float* __restrict__ escore,
                               const float* __restrict__ den,
                               const float* __restrict__ h1,
                               float* __restrict__ out) {
  int gid = blockIdx.x * blockDim.x + threadIdx.x;
  int wave = gid >> 5;
  int lane = gid & 31;
  int Etot = E + n;
  int nWaves = (gridDim.x * blockDim.x) >> 5;
  for (int e = wave; e < Etot; e += nWaves) {
    int s = (e < E) ? src[e] : (e - E);
    int d = (e < E) ? dst[e] : (e - E);
#pragma unroll
    for (int hd = 0; hd < 4; ++hd) {
      float alpha = escore[(size_t)e * 4 + hd] / den[d * 4 + hd];
      int c0 = hd * 64 + lane * 2;
      atomicAdd(&out[(size_t)d * 256 + c0], alpha * h1[(size_t)s * 256 + c0]);
      atomicAdd(&out[(size_t)d * 256 + c0 + 1],
                alpha * h1[(size_t)s * 256 + c0 + 1]);
    }
  }
}

// Pass C (layer 2, 1 head, 16 ch): thread per (edge, channel)
__global__ void gat_aggregate2(const int* __restrict__ src,
                               const int* __restrict__ dst, int E, int n,
                               const float* __restrict__ escore,
                               const float* __restrict__ den,
                               const float* __restrict__ h2,
                               float* __restrict__ out) {
  int gid = blockIdx.x * blockDim.x + threadIdx.x;
  int Etot = E + n;
  if (gid >= Etot * 16) return;
  int e = gid >> 4;
  int c = gid & 15;
  int s = (e < E) ? src[e] : (e - E);
  int d = (e < E) ? dst[e] : (e - E);
  float alpha = escore[e] / den[d];
  atomicAdd(&out[(size_t)d * 16 + c], alpha * h2[(size_t)s * 16 + c]);
}

// out1 = elu(agg + bias), elementwise
__global__ void gat_elu_bias(float* __restrict__ p,
                             const float* __restrict__ bias, size_t total,
                             int cols) {
  size_t i = (size_t)blockIdx.x * blockDim.x + threadIdx.x;
  size_t stride = (size_t)gridDim.x * blockDim.x;
  for (; i < total; i += stride) {
    float v = p[i] + bias[i % cols];
    p[i] = v > 0.f ? v : (__expf(v) - 1.f);
  }
}

// final: log_softmax over 16 classes per node (+ bias2)
__global__ void gat_logsoftmax16(const float* __restrict__ agg,
                                 const float* __restrict__ bias,
                                 float* __restrict__ out, int n) {
  int node = blockIdx.x * blockDim.x + threadIdx.x;
  if (node >= n) return;
  float v[16];
  float mx = -INFINITY;
#pragma unroll
  for (int c = 0; c < 16; ++c) {
    v[c] = agg[(size_t)node * 16 + c] + bias[c];
    mx = fmaxf(mx, v[c]);
  }
  float ssum = 0.f;
#pragma unroll
  for (int c = 0; c < 16; ++c) ssum += __expf(v[c] - mx);
  float ls = logf(ssum);
#pragma unroll
  for (int c = 0; c < 16; ++c) out[(size_t)node * 16 + c] = v[c] - mx - ls;
}

// ---------------------------------------------------------------------------
extern "C" void kernel_launch(void* const* d_in, const int* in_sizes, int n_in,
                              void* d_out, int out_size, void* d_ws,
                              size_t ws_size, hipStream_t stream) {
  const int F = 128, HC = 256, NCLS = 16;
  const int N = in_sizes[0] / F;
  const int E = in_sizes[1] / 2;
  const int Etot = E + N;

  const float* x   = (const float*)d_in[0];
  const int*   ei  = (const int*)d_in[1];
  const int*   srcIdx = ei;
  const int*   dstIdx = ei + E;
  const float* W1  = (const float*)d_in[2];
  const float* as1 = (const float*)d_in[3];
  const float* ad1 = (const float*)d_in[4];
  const float* b1  = (const float*)d_in[5];
  const float* W2  = (const float*)d_in[6];
  const float* as2 = (const float*)d_in[7];
  const float* ad2 = (const float*)d_in[8];
  const float* b2  = (const float*)d_in[9];
  float* out = (float*)d_out;

  // workspace carve-up
  float* w = (float*)d_ws;
  float* h1    = w; w += (size_t)N * HC;
  float* ssrc1 = w; w += (size_t)N * 4;
  float* sdst1 = w; w += (size_t)N * 4;
  float* m1    = w; w += (size_t)N * 4;
  float* den1  = w; w += (size_t)N * 4;
  float* e1    = w; w += (size_t)Etot * 4;
  float* out1  = w; w += (size_t)N * HC;
  float* h2    = w; w += (size_t)N * NCLS;
  float* ssrc2 = w; w += (size_t)N;
  float* sdst2 = w; w += (size_t)N;
  float* m2    = w; w += (size_t)N;
  float* den2  = w; w += (size_t)N;
  float* e2    = w; w += (size_t)Etot;
  float* agg2  = w; w += (size_t)N * NCLS;

  const int B = 256;
  auto blocks = [](size_t total, int b) { return (unsigned)((total + b - 1) / b); };

  // init accumulators (ws is poisoned by harness)
  gat_fill_f32<<<blocks((size_t)N * 4, B), B, 0, stream>>>(m1, -INFINITY, (size_t)N * 4);
  gat_fill_f32<<<blocks((size_t)N * 4, B), B, 0, stream>>>(den1, 0.f, (size_t)N * 4);
  gat_fill_f32<<<blocks((size_t)N * HC, B), B, 0, stream>>>(out1, 0.f, (size_t)N * HC);
  gat_fill_f32<<<blocks((size_t)N, B), B, 0, stream>>>(m2, -INFINITY, (size_t)N);
  gat_fill_f32<<<blocks((size_t)N, B), B, 0, stream>>>(den2, 0.f, (size_t)N);
  gat_fill_f32<<<blocks((size_t)N * NCLS, B), B, 0, stream>>>(agg2, 0.f, (size_t)N * NCLS);

  // ---- layer 1 ----
  {
    int tiles = (N / 16) * (HC / 16);            // 50000
    int wavesPerBlock = B / 32;                  // 8
    gat_wmma_gemm_f32<<<blocks(tiles, wavesPerBlock), B, 0, stream>>>(
        x, W1, h1, N, F, HC);
  }
  gat_attn_scores<<<blocks((size_t)N * 4, B), B, 0, stream>>>(
      h1, as1, ad1, ssrc1, sdst1, N, 4, 64);
  gat_edge_scores<<<blocks(Etot, B), B, 0, stream>>>(
      srcIdx, dstIdx, E, N, ssrc1, sdst1, 4, e1, m1);
  gat_edge_exp<<<blocks(Etot, B), B, 0, stream>>>(
      srcIdx, dstIdx, E, N, 4, e1, m1, den1);
  gat_aggregate1<<<blocks((size_t)Etot * 32, B), B, 0, stream>>>(
      srcIdx, dstIdx, E, N, e1, den1, h1, out1);
  gat_elu_bias<<<blocks((size_t)N * HC, B), B, 0, stream>>>(
      out1, b1, (size_t)N * HC, HC);

  // ---- layer 2 ----
  {
    int tiles = (N / 16) * (NCLS / 16);          // 3125
    int wavesPerBlock = B / 32;
    gat_wmma_gemm_f32<<<blocks(tiles, wavesPerBlock), B, 0, stream>>>(
        out1, W2, h2, N, HC, NCLS);
  }
  gat_attn_scores<<<blocks((size_t)N, B), B, 0, stream>>>(
      h2, as2, ad2, ssrc2, sdst2, N, 1, NCLS);
  gat_edge_scores<<<blocks(Etot, B), B, 0, stream>>>(
      srcIdx, dstIdx, E, N, ssrc2, sdst2, 1, e2, m2);
  gat_edge_exp<<<blocks(Etot, B), B, 0, stream>>>(
      srcIdx, dstIdx, E, N, 1, e2, m2, den2);
  gat_aggregate2<<<blocks((size_t)Etot * 16, B), B, 0, stream>>>(
      srcIdx, dstIdx, E, N, e2, den2, h2, agg2);

  gat_logsoftmax16<<<blocks((size_t)N, B), B, 0, stream>>>(agg2, b2, out, N);
}